// EncoderBlock_38139309588869
// MI455X (gfx1250) — compile-verified
//
#include <hip/hip_runtime.h>
#include <hip/hip_bf16.h>
#include <math.h>

typedef float v2f __attribute__((ext_vector_type(2)));
typedef float v8f __attribute__((ext_vector_type(8)));

namespace {
constexpr int kP     = 256;   // num patches
constexpr int kH     = 32;    // heads
constexpr int kDM    = 2048;  // d_model
constexpr int kHD    = 64;    // head dim
constexpr int kPATCH = 256;   // input feature dim

// workspace plan (float offsets): vr, vi, yr, yi — each (256 x 2048) f32 = 2MB
constexpr int VR_OFF = 0;
constexpr int VI_OFF = kP * kDM;
constexpr int YR_OFF = 2 * kP * kDM;
constexpr int YI_OFF = 3 * kP * kDM;
}

__device__ __forceinline__ v8f wmma_f32(v2f a, v2f b, v8f c) {
  // D = A(16x4, f32) * B(4x16, f32) + C(16x16, f32)
  return __builtin_amdgcn_wmma_f32_16x16x4_f32(false, a, false, b, (short)0, c,
                                               false, false);
}

// gfx1250 async copy: global -> LDS, 16 bytes per lane, tracked by ASYNCcnt
__device__ __forceinline__ void async_b128(unsigned int lds_off,
                                           const float* gaddr) {
  asm volatile("global_load_async_to_lds_b128 %0, %1, off"
               :
               : "v"(lds_off), "v"(gaddr)
               : "memory");
}
#define WAIT_ASYNC(n) asm volatile("s_wait_asynccnt " #n ::: "memory")

// ---------------------------------------------------------------------------
// Kernel 1: vr = x @ vR_k + vR_b ; vi = x @ vI_k + vI_b   (256x256)@(256x2048)
// One wave per 16x16 output tile position; A-fragment shared between R and I.
// ---------------------------------------------------------------------------
__global__ __launch_bounds__(256) void proj_kernel(
    const float* __restrict__ x, const float* __restrict__ vRk,
    const float* __restrict__ vRb, const float* __restrict__ vIk,
    const float* __restrict__ vIb, float* __restrict__ ws) {
  const int lane = threadIdx.x & 31;
  const int wave = (blockIdx.x * blockDim.x + threadIdx.x) >> 5;  // 0..2047
  const int half = lane >> 4;      // 0/1 -> K pair {0,1} or {2,3}
  const int l16  = lane & 15;
  const int tm   = wave >> 7;      // 0..15  (M tiles)
  const int tn   = wave & 127;     // 0..127 (N tiles)
  const int row  = tm * 16 + l16;  // A row for this lane
  const int col  = tn * 16 + l16;  // B/D column for this lane

  v8f cR = {};
  v8f cI = {};
  const float* xrow = x + row * kPATCH + 2 * half;
#pragma unroll 4
  for (int k0 = 0; k0 < kPATCH; k0 += 4) {
    v2f a;
    a.x = xrow[k0];
    a.y = xrow[k0 + 1];
    const int kb = k0 + 2 * half;
    v2f bR, bI;
    bR.x = vRk[kb * kDM + col];
    bR.y = vRk[(kb + 1) * kDM + col];
    bI.x = vIk[kb * kDM + col];
    bI.y = vIk[(kb + 1) * kDM + col];
    cR = wmma_f32(a, bR, cR);
    cI = wmma_f32(a, bI, cI);
  }
  const float br = vRb[col];
  const float bi = vIb[col];
#pragma unroll
  for (int r = 0; r < 8; ++r) {
    const int orow = tm * 16 + r + 8 * half;
    ws[VR_OFF + orow * kDM + col] = cR[r] + br;
    ws[VI_OFF + orow * kDM + col] = cI[r] + bi;
  }
}

// ---------------------------------------------------------------------------
// Kernel 2: per head h, complex circulant GEMM
//   y[s, h*64+d] = sum_q (JR+iJI)[(q+s)%256, h] * (vr+ivi)[q, h*64+d]
// A-matrix M_h[s][q] = J[(q+s)%256, h] gathered on the fly.
// ---------------------------------------------------------------------------
__global__ __launch_bounds__(256) void attn_kernel(
    const float* __restrict__ JR, const float* __restrict__ JI,
    float* __restrict__ ws) {
  const int lane = threadIdx.x & 31;
  const int wave = (blockIdx.x * blockDim.x + threadIdx.x) >> 5;  // 0..2047
  const int half = lane >> 4;
  const int l16  = lane & 15;
  const int h    = wave >> 6;                  // 0..31
  const int ts   = (wave >> 2) & 15;           // 0..15 (s tiles)
  const int td   = wave & 3;                   // 0..3  (d tiles)
  const int s    = ts * 16 + l16;              // A row (output position)
  const int d    = h * kHD + td * 16 + l16;    // column in (q, 2048) layout

  const float* vr = ws + VR_OFF;
  const float* vi = ws + VI_OFF;

  v8f cRe = {};
  v8f cIm = {};
#pragma unroll 4
  for (int k0 = 0; k0 < kP; k0 += 4) {
    const int q0 = k0 + 2 * half;
    const int p0 = (q0 + s) & (kP - 1);
    const int p1 = (q0 + 1 + s) & (kP - 1);
    v2f aR, aI, aIn;
    aR.x = JR[p0 * kH + h];
    aR.y = JR[p1 * kH + h];
    aI.x = JI[p0 * kH + h];
    aI.y = JI[p1 * kH + h];
    aIn.x = -aI.x;
    aIn.y = -aI.y;
    v2f bR, bI;
    bR.x = vr[q0 * kDM + d];
    bR.y = vr[(q0 + 1) * kDM + d];
    bI.x = vi[q0 * kDM + d];
    bI.y = vi[(q0 + 1) * kDM + d];
    cRe = wmma_f32(aR, bR, cRe);   // + MR @ vr
    cRe = wmma_f32(aIn, bI, cRe);  // - MI @ vi
    cIm = wmma_f32(aR, bI, cIm);   // + MR @ vi
    cIm = wmma_f32(aI, bR, cIm);   // + MI @ vr
  }
#pragma unroll
  for (int r = 0; r < 8; ++r) {
    const int orow = ts * 16 + r + 8 * half;
    ws[YR_OFF + orow * kDM + d] = cRe[r];
    ws[YI_OFF + orow * kDM + d] = cIm[r];
  }
}

// ---------------------------------------------------------------------------
// Complex log-cosh: sgn = sign(Re z); x = sgn*z;
// result = x + log(1 + exp(-2x)) - ln2   (complex exp/log)
// ---------------------------------------------------------------------------
__device__ __forceinline__ void log_cosh_c(float zr, float zi, float& outr,
                                           float& outi) {
  const float s = (__float_as_uint(zr) >> 31) ? -1.0f : 1.0f;
  const float xr = zr * s;
  const float xi = zi * s;
  const float t = expf(-2.0f * xr);          // xr >= 0 -> t in (0,1]
  float sn, cs;
  sincosf(2.0f * xi, &sn, &cs);
  const float wr = t * cs;                   // Re exp(-2x)
  const float wi = -t * sn;                  // Im exp(-2x)
  const float u = 1.0f + wr;
  const float lr = 0.5f * logf(u * u + wi * wi);
  const float li = atan2f(wi, u);
  outr = xr + lr - 0.69314718055994530942f;
  outi = xi + li;
}

// ---------------------------------------------------------------------------
// Kernel 3: out.re = yr@w0R - yi@w0I + bR ; out.im = yi@w0R + yr@w0I + bI
// Block = 8 waves = all 256 rows x 32-column strip. B panels (w0R/w0I) are
// double-buffered in LDS via async global->LDS B128 copies (ASYNCcnt).
// Each wave: 2 M-tiles x 2 N-tiles x {Re,Im} accumulators (8 x v8f).
// ---------------------------------------------------------------------------
__global__ __launch_bounds__(256) void out_kernel(
    const float* __restrict__ w0R, const float* __restrict__ w0Rb,
    const float* __restrict__ w0I, const float* __restrict__ w0Ib,
    const float* __restrict__ ws, float* __restrict__ out) {
  constexpr int KC = 64;                    // K-chunk depth
  __shared__ float sB[2][2][KC * 32];       // [buf][Re/Im][k][col], 32 KB

  const int t     = threadIdx.x;
  const int lane  = t & 31;
  const int w     = t >> 5;                 // wave 0..7 -> M-tiles {2w, 2w+1}
  const int half  = lane >> 4;
  const int l16   = lane & 15;
  const int strip = blockIdx.x * 32;        // 64 blocks cover N=2048

  // A row pointers (yr/yi), one per M-tile, pre-offset by the lane's K pair
  const float* yA[2][2];                    // [Re/Im][mt]
#pragma unroll
  for (int mt = 0; mt < 2; ++mt) {
    const int row = (w * 2 + mt) * 16 + l16;
    yA[0][mt] = ws + YR_OFF + row * kDM + 2 * half;
    yA[1][mt] = ws + YI_OFF + row * kDM + 2 * half;
  }

  // async-copy slot for this thread: 2 x B128 per matrix per chunk
  unsigned int ldsBase[2][2];               // [buf][Re/Im]
#pragma unroll
  for (int b = 0; b < 2; ++b) {
    ldsBase[b][0] = (unsigned int)(uintptr_t)&sB[b][0][0];
    ldsBase[b][1] = (unsigned int)(uintptr_t)&sB[b][1][0];
  }
  const int o0 = t * 16;                    // byte offset in 8KB panel
  const int o1 = t * 16 + 4096;
  const int r0 = o0 >> 7, c0 = (o0 & 127) >> 2;  // (row, col) of 4-float seg
  const int r1 = o1 >> 7, c1 = (o1 & 127) >> 2;

  v8f cRe[2][2] = {};
  v8f cIm[2][2] = {};

  // prime chunk 0 into buffer 0
  {
    const float* gR0 = w0R + (size_t)r0 * kDM + strip + c0;
    const float* gR1 = w0R + (size_t)r1 * kDM + strip + c1;
    const float* gI0 = w0I + (size_t)r0 * kDM + strip + c0;
    const float* gI1 = w0I + (size_t)r1 * kDM + strip + c1;
    async_b128(ldsBase[0][0] + o0, gR0);
    async_b128(ldsBase[0][0] + o1, gR1);
    async_b128(ldsBase[0][1] + o0, gI0);
    async_b128(ldsBase[0][1] + o1, gI1);
  }

  for (int ch = 0; ch < kDM / KC; ++ch) {
    if (ch + 1 < kDM / KC) {
      const int kn = (ch + 1) * KC;
      const int nb = (ch + 1) & 1;
      async_b128(ldsBase[nb][0] + o0, w0R + (size_t)(kn + r0) * kDM + strip + c0);
      async_b128(ldsBase[nb][0] + o1, w0R + (size_t)(kn + r1) * kDM + strip + c1);
      async_b128(ldsBase[nb][1] + o0, w0I + (size_t)(kn + r0) * kDM + strip + c0);
      async_b128(ldsBase[nb][1] + o1, w0I + (size_t)(kn + r1) * kDM + strip + c1);
      WAIT_ASYNC(0x4);                      // current chunk's 4 copies done
    } else {
      WAIT_ASYNC(0x0);
    }
    __syncthreads();                        // panel visible to all waves

    const float* bRe = &sB[ch & 1][0][0];
    const float* bIm = &sB[ch & 1][1][0];
    const int kgBase = ch * KC;
#pragma unroll 4
    for (int kk = 0; kk < KC; kk += 4) {
      const int kg = kgBase + kk;           // global K for A fragments
      const int kb = kk + 2 * half;         // local K row for B fragments
      v2f aR[2], aI[2], aIn[2];
#pragma unroll
      for (int mt = 0; mt < 2; ++mt) {
        aR[mt].x = yA[0][mt][kg];
        aR[mt].y = yA[0][mt][kg + 1];
        aI[mt].x = yA[1][mt][kg];
        aI[mt].y = yA[1][mt][kg + 1];
        aIn[mt].x = -aI[mt].x;
        aIn[mt].y = -aI[mt].y;
      }
      v2f bRf[2], bIf[2];
#pragma unroll
      for (int nt = 0; nt < 2; ++nt) {
        const int c = nt * 16 + l16;
        bRf[nt].x = bRe[kb * 32 + c];
        bRf[nt].y = bRe[(kb + 1) * 32 + c];
        bIf[nt].x = bIm[kb * 32 + c];
        bIf[nt].y = bIm[(kb + 1) * 32 + c];
      }
#pragma unroll
      for (int mt = 0; mt < 2; ++mt) {
#pragma unroll
        for (int nt = 0; nt < 2; ++nt) {
          cRe[mt][nt] = wmma_f32(aR[mt], bRf[nt], cRe[mt][nt]);   // + yr@w0R
          cRe[mt][nt] = wmma_f32(aIn[mt], bIf[nt], cRe[mt][nt]);  // - yi@w0I
          cIm[mt][nt] = wmma_f32(aI[mt], bRf[nt], cIm[mt][nt]);   // + yi@w0R
          cIm[mt][nt] = wmma_f32(aR[mt], bIf[nt], cIm[mt][nt]);   // + yr@w0I
        }
      }
    }
    __syncthreads();                        // done reading this buffer
  }

  // epilogue: bias + complex log-cosh, interleaved complex64 store
#pragma unroll
  for (int mt = 0; mt < 2; ++mt) {
#pragma unroll
    for (int nt = 0; nt < 2; ++nt) {
      const int col = strip + nt * 16 + l16;
      const float br = w0Rb[col];
      const float bi = w0Ib[col];
#pragma unroll
      for (int r = 0; r < 8; ++r) {
        const int orow = (w * 2 + mt) * 16 + r + 8 * half;
        float outr, outi;
        log_cosh_c(cRe[mt][nt][r] + br, cIm[mt][nt][r] + bi, outr, outi);
        out[(orow * kDM + col) * 2 + 0] = outr;
        out[(orow * kDM + col) * 2 + 1] = outi;
      }
    }
  }
}

extern "C" void kernel_launch(void* const* d_in, const int* in_sizes, int n_in,
                              void* d_out, int out_size, void* d_ws,
                              size_t ws_size, hipStream_t stream) {
  (void)in_sizes; (void)n_in; (void)out_size; (void)ws_size;
  const float* x    = (const float*)d_in[0];
  const float* vRk  = (const float*)d_in[1];
  const float* vRb  = (const float*)d_in[2];
  const float* vIk  = (const float*)d_in[3];
  const float* vIb  = (const float*)d_in[4];
  const float* JR   = (const float*)d_in[5];
  const float* JI   = (const float*)d_in[6];
  const float* w0Rk = (const float*)d_in[7];
  const float* w0Rb = (const float*)d_in[8];
  const float* w0Ik = (const float*)d_in[9];
  const float* w0Ib = (const float*)d_in[10];
  float* ws  = (float*)d_ws;
  float* out = (float*)d_out;

  proj_kernel<<<256, 256, 0, stream>>>(x, vRk, vRb, vIk, vIb, ws);
  attn_kernel<<<256, 256, 0, stream>>>(JR, JI, ws);
  out_kernel<<<64, 256, 0, stream>>>(w0Rk, w0Rb, w0Ik, w0Ib, ws, out);
}